// PADigitalTwin_26499948216470
// MI455X (gfx1250) — compile-verified
//
#include <hip/hip_runtime.h>
#include <math.h>

// Problem constants (fixed by the reference).
#define B_DIM   16
#define S_DIM   131072
#define MEM_M   5
#define HALO    (MEM_M - 1)                    // 4
#define OUT_N   (S_DIM - MEM_M + 1)            // 131068
#define TILE_N  2048
#define NTHR    256
#define SPT     (TILE_N / NTHR)                // 8 samples/thread, stride NTHR
#define TILES_X ((OUT_N + TILE_N - 1) / TILE_N) // 64
#define NPART   (TILES_X * B_DIM)              // 1024 block partials
#define LDS_FLOATS ((TILE_N + HALO) * 2)       // 4104 floats
#define LDS_BYTES  (LDS_FLOATS * 4)            // 16416 B
#define NB128      (LDS_BYTES / 16)            // 1026 b128 transfers per tile

typedef __attribute__((address_space(3))) float lds_f;
typedef __attribute__((ext_vector_type(2))) float v2f;
typedef __attribute__((ext_vector_type(8))) float v8f;

// ---- CDNA5 async global->LDS staging of one tile (+halo) -------------------
__device__ __forceinline__ void stage_tile_async(const float* __restrict__ x,
                                                 float* sh, int b,
                                                 long long n0, int t)
{
    unsigned ldsBase = (unsigned)(unsigned long long)(lds_f*)sh;
    const long long rowByte = ((long long)b * S_DIM + n0) * 8;
    const long long maxByte = (long long)B_DIM * S_DIM * 8 - 16; // clamp: never OOB
    const char* base = (const char*)x;
#pragma unroll
    for (int j = 0; j < 4; ++j) {
        long long off = ((long long)j * NTHR + t) * 16;
        long long gb  = rowByte + off;
        gb = gb < maxByte ? gb : maxByte;
        const char* g = base + gb;
        unsigned l = ldsBase + (unsigned)off;
        asm volatile("global_load_async_to_lds_b128 %0, %1, off"
                     :: "v"(l), "v"(g) : "memory");
    }
    if (t < (NB128 - 4 * NTHR)) {  // 2 tail transfers cover the halo bytes
        long long off = ((long long)4 * NTHR + t) * 16;
        long long gb  = rowByte + off;
        gb = gb < maxByte ? gb : maxByte;
        const char* g = base + gb;
        unsigned l = ldsBase + (unsigned)off;
        asm volatile("global_load_async_to_lds_b128 %0, %1, off"
                     :: "v"(l), "v"(g) : "memory");
    }
    asm volatile("s_wait_asynccnt 0" ::: "memory");  // my wave's ASYNCcnt drained
    __syncthreads();                                  // all waves' LDS writes visible
}

// ---- coefficients: uniform indices -> scalar (SMEM) loads ------------------
__device__ __forceinline__ void load_coeffs(const float* __restrict__ cr,
                                            const float* __restrict__ ci,
                                            float cm_r[MEM_M][4],
                                            float cm_i[MEM_M][4])
{
#pragma unroll
    for (int k = 0; k < 4; ++k)
#pragma unroll
        for (int m = 0; m < MEM_M; ++m) {
            cm_r[m][k] = cr[k * MEM_M + m];
            cm_i[m][k] = ci[k * MEM_M + m];
        }
}

// ---- Volterra sample: 5 taps x (2 + 3 + 3 + 4) FMA = 60 FMA ---------------
__device__ __forceinline__ void volterra_eval(const float* sh, int l,
                                              const float cm_r[MEM_M][4],
                                              const float cm_i[MEM_M][4],
                                              float& yr, float& yi)
{
    yr = 0.0f; yi = 0.0f;
#pragma unroll
    for (int m = 0; m < MEM_M; ++m) {
        int idx = l + (HALO - m);                // coeff m pairs with x[n + 4 - m]
        float Ii = sh[2 * idx + 0];
        float Qq = sh[2 * idx + 1];
        float a2 = fmaf(Ii, Ii, Qq * Qq);
        float wr = fmaf(fmaf(fmaf(cm_r[m][3], a2, cm_r[m][2]), a2, cm_r[m][1]), a2, cm_r[m][0]);
        float wi = fmaf(fmaf(fmaf(cm_i[m][3], a2, cm_i[m][2]), a2, cm_i[m][1]), a2, cm_i[m][0]);
        yr = fmaf(wr, Ii, fmaf(-wi, Qq, yr));
        yi = fmaf(wr, Qq, fmaf( wi, Ii, yi));
    }
}

// ---- pass 1: per-block signal-power partials (deterministic) ---------------
__global__ __launch_bounds__(NTHR) void pa_pass1(const float* __restrict__ x,
                                                 const float* __restrict__ cr,
                                                 const float* __restrict__ ci,
                                                 float* __restrict__ partials)
{
    __shared__ __align__(16) float sh[LDS_FLOATS];
    __shared__ float red[NTHR];
    const int t = threadIdx.x;
    const int b = blockIdx.y;
    const long long n0 = (long long)blockIdx.x * TILE_N;

    float cm_r[MEM_M][4], cm_i[MEM_M][4];
    load_coeffs(cr, ci, cm_r, cm_i);
    stage_tile_async(x, sh, b, n0, t);

    float acc = 0.0f;
#pragma unroll
    for (int j = 0; j < SPT; ++j) {
        int l = t + j * NTHR;                    // 8B lane stride in LDS: bank-clean
        if (n0 + l < OUT_N) {
            float yr, yi;
            volterra_eval(sh, l, cm_r, cm_i, yr, yi);
            acc = fmaf(yr, yr, fmaf(yi, yi, acc));
        }
    }
    red[t] = acc;
    __syncthreads();
#pragma unroll
    for (int s = NTHR / 2; s > 0; s >>= 1) {
        if (t < s) red[t] += red[t + s];
        __syncthreads();
    }
    if (t == 0) partials[blockIdx.y * gridDim.x + blockIdx.x] = red[0];
}

// ---- final reduce: one wave, V_WMMA_F32_16X16X4_F32 ------------------------
// D = A(16x4) x ones(4x16) + C replicates every A-row-sum across 16 columns.
// Summing all 256 D entries therefore yields 16 * (sum of all A inputs),
// independent of the A/B/D VGPR layouts (only "B is all ones" matters).
// 16 chained WMMAs consume all 1024 partials in a fixed, deterministic order.
__global__ __launch_bounds__(32) void pa_reduce(const float* __restrict__ partials,
                                                float* __restrict__ sumbuf)
{
    const int t = threadIdx.x;                   // exactly one wave32, EXEC all-1s
    const v2f ones = {1.0f, 1.0f};
    v8f c = {};
#pragma unroll
    for (int it = 0; it < NPART / 64; ++it) {    // 16 iterations x 64 values
        v2f a;
        a.x = partials[it * 64 + 2 * t + 0];
        a.y = partials[it * 64 + 2 * t + 1];
        c = __builtin_amdgcn_wmma_f32_16x16x4_f32(false, a, false, ones,
                                                  (short)0, c, false, false);
    }
    float lane = (c[0] + c[1]) + (c[2] + c[3]) + ((c[4] + c[5]) + (c[6] + c[7]));
    __shared__ float red[32];
    red[t] = lane;
    __syncthreads();
#pragma unroll
    for (int k = 16; k > 0; k >>= 1) {
        if (t < k) red[t] += red[t + k];
        __syncthreads();
    }
    if (t == 0) sumbuf[0] = red[0] * (1.0f / 16.0f);   // undo 16x replication
}

// ---- pass 2: recompute (x hits L2), rotate, add noise, store ---------------
__global__ __launch_bounds__(NTHR) void pa_pass2(const float* __restrict__ x,
                                                 const float* __restrict__ cr,
                                                 const float* __restrict__ ci,
                                                 const float* __restrict__ awgn,
                                                 const float* __restrict__ phase,
                                                 const float* __restrict__ sumbuf,
                                                 float2* __restrict__ out)
{
    __shared__ __align__(16) float sh[LDS_FLOATS];
    const int t = threadIdx.x;
    const int b = blockIdx.y;
    const long long n0 = (long long)blockIdx.x * TILE_N;

    float cm_r[MEM_M][4], cm_i[MEM_M][4];
    load_coeffs(cr, ci, cm_r, cm_i);

    // signal_power = mean(yr^2+yi^2) * 0.5 ; noise_std = sqrt(sp * 1e-6)
    const float mean = sumbuf[0] * (1.0f / ((float)B_DIM * (float)OUT_N));
    const float nstd = sqrtf(mean * 0.5f * 1.0e-6f);
    const float PH = 0.008726646259971648f;     // 0.5 deg in radians

    stage_tile_async(x, sh, b, n0, t);

#pragma unroll
    for (int j = 0; j < SPT; ++j) {
        int l = t + j * NTHR;
        long long n = n0 + l;
        if (n < OUT_N) {
            float yr, yi;
            volterra_eval(sh, l, cm_r, cm_i, yr, yi);
            long long gi = (long long)b * OUT_N + n;
            float pr = phase[gi] * PH;
            float sn, cs;
            sincosf(pr, &sn, &cs);
            float yr2 = yr * cs - yi * sn;
            float yi2 = yr * sn + yi * cs;
            float2 o;
            o.x = fmaf(awgn[2 * gi + 0], nstd, yr2);
            o.y = fmaf(awgn[2 * gi + 1], nstd, yi2);
            out[gi] = o;
        }
    }
}

extern "C" void kernel_launch(void* const* d_in, const int* in_sizes, int n_in,
                              void* d_out, int out_size, void* d_ws, size_t ws_size,
                              hipStream_t stream) {
    const float* x     = (const float*)d_in[0];
    const float* cr    = (const float*)d_in[1];
    const float* ci    = (const float*)d_in[2];
    const float* awgn  = (const float*)d_in[3];
    const float* phase = (const float*)d_in[4];

    float* sumbuf   = (float*)d_ws;       // ws[0]  : global power sum
    float* partials = sumbuf + 1;         // ws[1..]: 1024 block partials

    dim3 grid(TILES_X, B_DIM);
    pa_pass1 <<<grid, NTHR, 0, stream>>>(x, cr, ci, partials);
    pa_reduce<<<1,    32,   0, stream>>>(partials, sumbuf);
    pa_pass2 <<<grid, NTHR, 0, stream>>>(x, cr, ci, awgn, phase, sumbuf,
                                         (float2*)d_out);
}